// KAN_13675175870681
// MI455X (gfx1250) — compile-verified
//
#include <hip/hip_runtime.h>
#include <hip/hip_bf16.h>

typedef __attribute__((ext_vector_type(2))) float v2f;
typedef __attribute__((ext_vector_type(8))) float v8f;
typedef int v4i __attribute__((vector_size(16)));

// Problem constants (match reference)
constexpr int BATCH = 512;
constexpr int DIM   = 256;    // inDim == outDim == 256 for both layers
constexpr int NCPS  = 16;
constexpr int PSLOT = 20;     // 18 padded-cp slots + silu + zero pad
constexpr int KTOT  = DIM * PSLOT;   // 5120
constexpr float DXK = 1.0f / 7.0f;   // knot spacing: dx = 2/(nCps-2) = 2/14

// ---------------------------------------------------------------------------
// CDNA5 async global->LDS copy (ASYNCcnt path), with inline-asm fallback.
// Builtin signature (from probe): (v4i addrspace(1)*, v4i addrspace(3)*, Ii, Ii)
// ---------------------------------------------------------------------------
__device__ __forceinline__ void async_copy16(const float* g, void* l) {
#if defined(__HIP_DEVICE_COMPILE__)
#if __has_builtin(__builtin_amdgcn_global_load_async_to_lds_b128)
    __builtin_amdgcn_global_load_async_to_lds_b128(
        (__attribute__((address_space(1))) v4i*)g,
        (__attribute__((address_space(3))) v4i*)l, 0, 0);
#else
    asm volatile("global_load_async_to_lds_b128 %0, %1, off"
                 :: "v"((unsigned)(unsigned long long)l), "v"(g)
                 : "memory");
#endif
#endif
}

__device__ __forceinline__ void wait_async0() {
#if defined(__HIP_DEVICE_COMPILE__)
#if __has_builtin(__builtin_amdgcn_s_wait_asynccnt)
    __builtin_amdgcn_s_wait_asynccnt(0);
#else
    asm volatile("s_wait_asynccnt 0x0" ::: "memory");
#endif
#endif
}

// ---------------------------------------------------------------------------
// Kernel 1: Wc[i*20+p, o] = w0[i,o]*padded_cp[i,p,o] (p<18), w1[i,o] (p=18), 0 (p=19)
// cps: (DIM, DIM, NCPS), w: (2, DIM, DIM, 1)
// ---------------------------------------------------------------------------
__global__ __launch_bounds__(256) void kan_build_wc(const float* __restrict__ cps,
                                                    const float* __restrict__ w,
                                                    float* __restrict__ Wc) {
    const int i = blockIdx.x;
    const int o = threadIdx.x;
    const float w0v = w[(size_t)i * DIM + o];
    const float w1v = w[(size_t)DIM * DIM + (size_t)i * DIM + o];

    float c[NCPS];
    const float4* cp4 = (const float4*)(cps + ((size_t)i * DIM + o) * NCPS);
#pragma unroll
    for (int q = 0; q < NCPS / 4; ++q) {
        float4 v = cp4[q];
        c[4 * q + 0] = v.x; c[4 * q + 1] = v.y;
        c[4 * q + 2] = v.z; c[4 * q + 3] = v.w;
    }

    float* dst = Wc + (size_t)i * PSLOT * DIM + o;
#pragma unroll
    for (int p = 0; p < NCPS; ++p) dst[(size_t)p * DIM] = w0v * c[p];
    const float last = w0v * c[NCPS - 1];
    dst[(size_t)16 * DIM] = last;        // padded cp (repeat last, k-1 = 2 times)
    dst[(size_t)17 * DIM] = last;
    dst[(size_t)18 * DIM] = w1v;         // silu weight
    dst[(size_t)19 * DIM] = 0.0f;        // pad
}

// ---------------------------------------------------------------------------
// Kernel 2: Phi[b, i*20+p] — cubic B-spline basis (4 nonzero of 18) + silu + 0
// Interval from clamped x; recursion uses raw x (matches reference alpha).
// ---------------------------------------------------------------------------
__global__ __launch_bounds__(256) void kan_phi(const float* __restrict__ X,
                                               float* __restrict__ Phi) {
    const int b = blockIdx.x;
    const int i = threadIdx.x;
    const float x  = X[(size_t)b * DIM + i];
    const float xc = fminf(fmaxf(x, -1.0f), 1.0f);

    int left = 3 + (int)floorf((xc + 1.0f) * 7.0f);   // t[a] = -1 + (a-3)*dx
    left = max(3, min(17, left));

    // Cox–de Boor basis functions (Piegl & Tiller A2.2), k = 3
    float Nb[4], lv[4], rv[4];
    Nb[0] = 1.0f;
#pragma unroll
    for (int j = 1; j <= 3; ++j) {
        lv[j] = x - (-1.0f + (float)(left + 1 - j - 3) * DXK);
        rv[j] = (-1.0f + (float)(left + j - 3) * DXK) - x;
        float saved = 0.0f;
#pragma unroll
        for (int r = 0; r < j; ++r) {
            const float temp = Nb[r] / (rv[r + 1] + lv[j - r]);
            Nb[r] = saved + rv[r + 1] * temp;
            saved = lv[j - r] * temp;
        }
        Nb[j] = saved;
    }

    // Branch-free scatter of the 4 coefficients into 18 dense slots
    float out[PSLOT];
#pragma unroll
    for (int p = 0; p < 18; ++p) {
        float v = 0.0f;
#pragma unroll
        for (int m = 0; m < 4; ++m) v = (p == left - 3 + m) ? Nb[m] : v;
        out[p] = v;
    }
    out[18] = x * (1.0f / (1.0f + __expf(-x)));   // silu
    out[19] = 0.0f;

    float4* dst = (float4*)(Phi + (size_t)b * KTOT + i * PSLOT);
#pragma unroll
    for (int q = 0; q < PSLOT / 4; ++q)
        dst[q] = make_float4(out[4 * q], out[4 * q + 1], out[4 * q + 2], out[4 * q + 3]);
}

// ---------------------------------------------------------------------------
// Kernel 3: C[512x256] = Phi[512x5120] x Wc[5120x256], f32 WMMA 16x16x4
// Block: 64x64 tile, 256 threads = 8 waves; wave -> 16(M) x 32(N), 2 accums.
// A tile staged via CDNA5 async global->LDS DMA; BK=64 (80 outer iterations).
// ---------------------------------------------------------------------------
constexpr int BM = 64, BN = 64, BK = 64;
constexpr int SAPAD = BK + 4;   // 68-word rows: 16B-aligned async dests,
                                // stride = 4 mod 64 banks -> conflict-free cols
constexpr int SBPAD = BN + 4;   // 68-word rows

__global__ __launch_bounds__(256) void kan_gemm(const float* __restrict__ A,
                                                const float* __restrict__ Bm,
                                                float* __restrict__ C) {
    __shared__ float sA[BM][SAPAD];
    __shared__ float sB[BK][SBPAD];

    const int tid  = threadIdx.x;
    const int wave = tid >> 5;
    const int lane = tid & 31;
    const int lh   = lane >> 4;       // half-wave select (K split / M split)
    const int ll   = lane & 15;
    const int wm   = wave & 3;        // 4 waves along M  -> 16 rows each
    const int wn   = wave >> 2;       // 2 waves along N  -> 32 cols each
    const int bm0  = blockIdx.x * BM;
    const int bn0  = blockIdx.y * BN;

    v8f acc0 = {}; v8f acc1 = {};

    for (int kt = 0; kt < KTOT; kt += BK) {
        // --- Stage A tile (64x64) via async DMA: 1024 x b128, 4 per thread ---
        {
            int idx = tid;
#pragma unroll
            for (int it = 0; it < 4; ++it) {
                const int r = idx >> 4, seg = (idx & 15) << 2;
                async_copy16(A + (size_t)(bm0 + r) * KTOT + kt + seg,
                             (void*)&sA[r][seg]);
                idx += 256;
            }
        }
        // --- Stage B tile (64x64) via regular vectorized loads ---
        {
            int idxb = tid;
#pragma unroll
            for (int it = 0; it < 4; ++it) {
                const int r = idxb >> 4, c4 = (idxb & 15) << 2;
                const float4 v = *(const float4*)(Bm + (size_t)(kt + r) * DIM + bn0 + c4);
                sB[r][c4 + 0] = v.x; sB[r][c4 + 1] = v.y;
                sB[r][c4 + 2] = v.z; sB[r][c4 + 3] = v.w;
                idxb += 256;
            }
            if (kt + BK < KTOT)   // prefetch next B tile into cache
                __builtin_prefetch(Bm + (size_t)(kt + BK + (tid >> 4)) * DIM +
                                       bn0 + ((tid & 15) << 2), 0, 1);
        }
        wait_async0();            // A-tile DMA complete (this wave)
        __syncthreads();          // all waves' LDS writes visible

#pragma unroll
        for (int kk = 0; kk < BK; kk += 4) {
            // A frag 16x4: lanes 0-15 K={0,1}, lanes 16-31 K={2,3} (ISA layout)
            v2f a;
            a.x = sA[wm * 16 + ll][kk + lh * 2 + 0];
            a.y = sA[wm * 16 + ll][kk + lh * 2 + 1];
            // B frag 4x16: V0 rows K=0/K=2, V1 rows K=1/K=3, N striped across lanes
            const int n0 = wn * 32 + ll;
            v2f b0, b1;
            b0.x = sB[kk + lh * 2 + 0][n0];
            b0.y = sB[kk + lh * 2 + 1][n0];
            b1.x = sB[kk + lh * 2 + 0][n0 + 16];
            b1.y = sB[kk + lh * 2 + 1][n0 + 16];

            acc0 = __builtin_amdgcn_wmma_f32_16x16x4_f32(false, a, false, b0,
                                                         (short)0, acc0, false, false);
            acc1 = __builtin_amdgcn_wmma_f32_16x16x4_f32(false, a, false, b1,
                                                         (short)0, acc1, false, false);
        }
        __syncthreads();
    }

    // C/D layout: VGPR v -> M = v (lanes 0-15) / M = 8+v (lanes 16-31)
    const int row0 = bm0 + wm * 16 + lh * 8;
    const int col  = bn0 + wn * 32 + ll;
#pragma unroll
    for (int v = 0; v < 8; ++v) {
        C[(size_t)(row0 + v) * DIM + col]      = acc0[v];
        C[(size_t)(row0 + v) * DIM + col + 16] = acc1[v];
    }
}

// ---------------------------------------------------------------------------
extern "C" void kernel_launch(void* const* d_in, const int* in_sizes, int n_in,
                              void* d_out, int out_size, void* d_ws, size_t ws_size,
                              hipStream_t stream) {
    const float* x    = (const float*)d_in[0];   // (512, 256)
    const float* cps0 = (const float*)d_in[1];   // (256, 256, 16)
    const float* w0   = (const float*)d_in[2];   // (2, 256, 256, 1)
    const float* cps1 = (const float*)d_in[3];
    const float* w1   = (const float*)d_in[4];
    float* out = (float*)d_out;                  // (512, 256) f32

    char* ws = (char*)d_ws;
    float* Phi = (float*)(ws);                                        // 512*5120 f32
    float* Wc  = (float*)(ws + (size_t)BATCH * KTOT * sizeof(float)); // 5120*256 f32
    float* h   = (float*)(ws + (size_t)BATCH * KTOT * sizeof(float)
                             + (size_t)KTOT * DIM * sizeof(float));   // 512*256 f32

    const dim3 ggrid(BATCH / BM, DIM / BN);

    // Layer 0: x -> h
    kan_build_wc<<<DIM, DIM, 0, stream>>>(cps0, w0, Wc);
    kan_phi<<<BATCH, DIM, 0, stream>>>(x, Phi);
    kan_gemm<<<ggrid, 256, 0, stream>>>(Phi, Wc, h);

    // Layer 1: h -> out
    kan_build_wc<<<DIM, DIM, 0, stream>>>(cps1, w1, Wc);
    kan_phi<<<BATCH, DIM, 0, stream>>>(h, Phi);
    kan_gemm<<<ggrid, 256, 0, stream>>>(Phi, Wc, out);
}